// EmbeddingLayer_3736621548057
// MI455X (gfx1250) — compile-verified
//
#include <hip/hip_runtime.h>

#define N_NODES   50000
#define N_EDGES   800000
#define ATOM_SIZE 95
#define AE        128
#define BE        128
#define BOND_IN   64
#define EPS       1e-3f

typedef __attribute__((ext_vector_type(2))) float v2f;
typedef __attribute__((ext_vector_type(4))) float v4f;
typedef __attribute__((ext_vector_type(8))) float v8f;

// ---- workspace layout (floats) -------------------------------------------
// W12  : fused (W1 * s1) @ W2[128:256]            [64][128]
// TCI  : atom_table @ W2[0:128]   per atom TYPE   [95][128]
// TCJ  : atom_table @ W2[256:384] per atom TYPE   [95][128]
// CVEC : t1 @ W2[128:256] + b2 - mean2            [128]
// S2   : gamma2 * rsqrt(var2+eps)                 [128]
// B2C  : beta2 copy                               [128]
#define WS_W12   0
#define WS_TCI   8192
#define WS_TCJ   20352
#define WS_CVEC  32512
#define WS_S2    32640
#define WS_B2C   32768
#define WS_TOTAL 32896   // floats -> 131584 bytes, fits 320KB WGP LDS (2 blk/WGP)

// ---------------------------------------------------------------------------
// Kernel 1: fold BN1/BN2 into small fused tables (trivial FLOPs, scalar ok)
// ---------------------------------------------------------------------------
__global__ __launch_bounds__(256) void mpnn_prep_kernel(
    const float* __restrict__ atom_table,
    const float* __restrict__ W1,     const float* __restrict__ b1,
    const float* __restrict__ gamma1, const float* __restrict__ beta1,
    const float* __restrict__ mean1,  const float* __restrict__ var1,
    const float* __restrict__ W2,     const float* __restrict__ b2,
    const float* __restrict__ gamma2, const float* __restrict__ beta2,
    const float* __restrict__ mean2,  const float* __restrict__ var2,
    float* __restrict__ ws)
{
    const int tid = blockIdx.x * 256 + threadIdx.x;
    if (tid >= WS_TOTAL) return;

    if (tid < WS_TCI) {                       // W12[c][k]
        const int c = tid >> 7, k = tid & 127;
        float acc = 0.f;
        for (int m = 0; m < 128; ++m) {
            const float s1 = gamma1[m] * rsqrtf(var1[m] + EPS);
            acc += W1[c * 128 + m] * s1 * W2[(128 + m) * 128 + k];
        }
        ws[tid] = acc;
    } else if (tid < WS_TCJ) {                // TCI[a][k] = table @ W2a
        const int i = tid - WS_TCI, a = i >> 7, k = i & 127;
        float acc = 0.f;
        for (int m = 0; m < 128; ++m)
            acc += atom_table[a * 128 + m] * W2[m * 128 + k];
        ws[tid] = acc;
    } else if (tid < WS_CVEC) {               // TCJ[a][k] = table @ W2c
        const int i = tid - WS_TCJ, a = i >> 7, k = i & 127;
        float acc = 0.f;
        for (int m = 0; m < 128; ++m)
            acc += atom_table[a * 128 + m] * W2[(256 + m) * 128 + k];
        ws[tid] = acc;
    } else if (tid < WS_S2) {                 // cvec[k]
        const int k = tid - WS_CVEC;
        float acc = b2[k] - mean2[k];
        for (int m = 0; m < 128; ++m) {
            const float s1 = gamma1[m] * rsqrtf(var1[m] + EPS);
            const float t1 = (b1[m] - mean1[m]) * s1 + beta1[m];
            acc += t1 * W2[(128 + m) * 128 + k];
        }
        ws[tid] = acc;
    } else if (tid < WS_B2C) {                // s2[k]
        const int k = tid - WS_S2;
        ws[tid] = gamma2[k] * rsqrtf(var2[k] + EPS);
    } else {                                  // beta2 copy
        ws[tid] = beta2[tid - WS_B2C];
    }
}

// ---------------------------------------------------------------------------
// Kernel 2: atom_embedding output = gather rows of the 95x128 table (float4)
// ---------------------------------------------------------------------------
__global__ __launch_bounds__(256) void mpnn_atom_embed_kernel(
    const int*   __restrict__ atom_features,
    const float* __restrict__ atom_table,
    float*       __restrict__ out1)
{
    const int tid = blockIdx.x * 256 + threadIdx.x;    // one float4 each
    if (tid >= N_NODES * 32) return;
    const int n = tid >> 5;
    const int q = (tid & 31) << 2;
    const v4f val = *(const v4f*)(atom_table + atom_features[n] * 128 + q);
    *(v4f*)(out1 + (size_t)n * 128 + q) = val;
}

// ---------------------------------------------------------------------------
// Kernel 3: fused edge update via fp32 WMMA (16x16x4), LDS-resident tables
//   out2[e] = (bf[e] @ W12 + TCI[type(i[e])] + TCJ[type(j[e])] + cvec)*s2 + b2c
// ---------------------------------------------------------------------------
#define EDGE_BLOCKS 1024
#define NTILES (N_EDGES / 16)

__global__ __launch_bounds__(256) void mpnn_edge_wmma_kernel(
    const float* __restrict__ bond,      // [N_EDGES][64]
    const int*   __restrict__ ii,
    const int*   __restrict__ jj,
    const int*   __restrict__ atype,     // atom_features
    const float* __restrict__ ws,
    float*       __restrict__ out2)      // [N_EDGES][128]
{
    extern __shared__ float lds[];
    for (int i = threadIdx.x; i < WS_TOTAL; i += 256) lds[i] = ws[i];
    __syncthreads();

    const float* ldsW12 = lds + WS_W12;
    const float* ldsTCI = lds + WS_TCI;
    const float* ldsTCJ = lds + WS_TCJ;
    const float* ldsCV  = lds + WS_CVEC;
    const float* ldsS2  = lds + WS_S2;
    const float* ldsB2  = lds + WS_B2C;

    const int lane    = threadIdx.x & 31;
    const int n16     = lane & 15;        // column / A-row within halfwave
    const int khalf   = lane >> 4;        // which K pair (ISA f32 A/B layout)
    const int wid     = (blockIdx.x << 3) + (threadIdx.x >> 5);
    const int wstride = gridDim.x << 3;

    for (int tile = wid; tile < NTILES; tile += wstride) {
        const int ebase = tile << 4;      // 16 edges per wave-tile

        // A fragments: lane holds bf[ebase+n16][4t + 2*khalf + {0,1}]
        const float* arow = bond + (size_t)(ebase + n16) * 64 + (khalf << 1);
        v2f a[16];
#pragma unroll
        for (int t = 0; t < 16; ++t)
            a[t] = *(const v2f*)(arow + (t << 2));

        // per-edge gather offsets (edge -> node -> atom type, L2 resident)
        int offI[8], offJ[8];
#pragma unroll
        for (int v = 0; v < 8; ++v) {
            const int e = ebase + v + (khalf << 3);
            offI[v] = (atype[ii[e]] << 7) + n16;
            offJ[v] = (atype[jj[e]] << 7) + n16;
        }

        float* outp = out2 + (size_t)ebase * 128 + n16;

        for (int nb = 0; nb < 8; ++nb) {  // 8 column blocks of 16 -> 128 outs
            const int n0 = nb << 4;
            // B fragment base: row (2*khalf), column n0+n16 of W12[64][128]
            const float* bcol = ldsW12 + (khalf << 8) + n0 + n16;

            v8f acc = {};
#pragma unroll
            for (int t = 0; t < 16; ++t) {       // K = 64 in steps of 4
                v2f b;
                b.x = bcol[(t << 9)];            // row 4t + 2*khalf
                b.y = bcol[(t << 9) + 128];      // row 4t + 2*khalf + 1
                acc = __builtin_amdgcn_wmma_f32_16x16x4_f32(
                          false, a[t], false, b, (short)0, acc, false, false);
            }

            const float cv = ldsCV[n0 + n16];
            const float sv = ldsS2[n0 + n16];
            const float bv = ldsB2[n0 + n16];
#pragma unroll
            for (int v = 0; v < 8; ++v) {        // C/D layout: M = v + 8*khalf
                const float o = (acc[v] + ldsTCI[offI[v] + n0]
                                        + ldsTCJ[offJ[v] + n0] + cv) * sv + bv;
                outp[(size_t)(v + (khalf << 3)) * 128 + n0] = o;
            }
        }
    }
}

// ---------------------------------------------------------------------------
extern "C" void kernel_launch(void* const* d_in, const int* in_sizes, int n_in,
                              void* d_out, int out_size, void* d_ws, size_t ws_size,
                              hipStream_t stream)
{
    const int*   atom_features = (const int*)  d_in[0];
    const float* bond_features = (const float*)d_in[1];
    const int*   indices_i     = (const int*)  d_in[2];
    const int*   indices_j     = (const int*)  d_in[3];
    const float* atom_table    = (const float*)d_in[4];
    const float* W1            = (const float*)d_in[5];
    const float* b1            = (const float*)d_in[6];
    const float* gamma1        = (const float*)d_in[7];
    const float* beta1         = (const float*)d_in[8];
    const float* mean1         = (const float*)d_in[9];
    const float* var1          = (const float*)d_in[10];
    const float* W2            = (const float*)d_in[11];
    const float* b2            = (const float*)d_in[12];
    const float* gamma2        = (const float*)d_in[13];
    const float* beta2         = (const float*)d_in[14];
    const float* mean2         = (const float*)d_in[15];
    const float* var2          = (const float*)d_in[16];

    float* ws   = (float*)d_ws;
    float* out1 = (float*)d_out;                       // [N_NODES][128]
    float* out2 = out1 + (size_t)N_NODES * AE;         // [N_EDGES][128]

    mpnn_prep_kernel<<<(WS_TOTAL + 255) / 256, 256, 0, stream>>>(
        atom_table, W1, b1, gamma1, beta1, mean1, var1,
        W2, b2, gamma2, beta2, mean2, var2, ws);

    mpnn_atom_embed_kernel<<<(N_NODES * 32 + 255) / 256, 256, 0, stream>>>(
        atom_features, atom_table, out1);

    mpnn_edge_wmma_kernel<<<EDGE_BLOCKS, 256, WS_TOTAL * sizeof(float), stream>>>(
        bond_features, indices_i, indices_j, atom_features, ws, out2);
}